// FlaxPositionWiseMoELayer_91268055040497
// MI455X (gfx1250) — compile-verified
//
#include <hip/hip_runtime.h>

typedef float v2f __attribute__((ext_vector_type(2)));
typedef float v8f __attribute__((ext_vector_type(8)));

#define SS 2048
#define MM 1024
#define HH 2048
#define EE 64
#define GG 4
// capacity C = 64 -> final scale

__device__ __forceinline__ v8f wmma4(v2f a, v2f b, v8f c) {
  // D = A(16x4,f32) * B(4x16,f32) + C(16x16,f32)
  return __builtin_amdgcn_wmma_f32_16x16x4_f32(false, a, false, b, (short)0, c,
                                               false, false);
}

// ---------------- Kernel 1: gates[8192,64] = x[8192,1024] @ wg[1024,64] ----
__global__ __launch_bounds__(256) void k_gates(const float* __restrict__ x,
                                               const float* __restrict__ wg,
                                               float* __restrict__ gates) {
  const int lane = threadIdx.x & 31;
  const int wave = threadIdx.x >> 5;
  const int tile = blockIdx.x * 8 + wave;     // 512 row-tiles * 4 col-tiles
  const int rowTile = tile >> 2;
  const int colTile = tile & 3;
  const int half = lane >> 4;
  const int l15  = lane & 15;
  const int row  = rowTile * 16 + l15;        // A row (s index)
  const int col  = colTile * 16 + l15;        // B col (e index)
  v8f c = {};
#pragma unroll 4
  for (int k = 0; k < MM; k += 4) {
    const int ka = k + half * 2;
    v2f a = *(const v2f*)(x + row * MM + ka);  // contiguous K pair
    v2f b;
    b[0] = wg[ka * EE + col];
    b[1] = wg[(ka + 1) * EE + col];
    c = wmma4(a, b, c);
  }
  const int rbase = half * 8;
#pragma unroll
  for (int r = 0; r < 8; ++r)
    gates[(rowTile * 16 + rbase + r) * EE + col] = c[r];
}

// ------------- Kernel 2: D[g][64,1024] = gates_g^T[64,2048] @ x_g[2048,1024]
__global__ __launch_bounds__(256) void k_dispatch(const float* __restrict__ gates,
                                                  const float* __restrict__ x,
                                                  float* __restrict__ D) {
  const int lane = threadIdx.x & 31;
  const int wave = threadIdx.x >> 5;
  const int tile = blockIdx.x * 8 + wave;     // 4g * 4eTiles * 64mTiles = 1024
  const int g     = tile >> 8;
  const int eTile = (tile >> 6) & 3;
  const int mTile = tile & 63;
  const int half = lane >> 4;
  const int l15  = lane & 15;
  const int erow = eTile * 16 + l15;          // A row (expert)
  const int mcol = mTile * 16 + l15;          // B col (m)
  v8f c = {};
#pragma unroll 2
  for (int k = 0; k < SS; k += 4) {
    const int ka = k + half * 2;
    v2f a, b;
    a[0] = gates[(g * SS + ka) * EE + erow];      // A[e,s] = gates[g,s,e]
    a[1] = gates[(g * SS + ka + 1) * EE + erow];
    b[0] = x[(g * SS + ka) * MM + mcol];
    b[1] = x[(g * SS + ka + 1) * MM + mcol];
    c = wmma4(a, b, c);
  }
  const int rbase = half * 8;
#pragma unroll
  for (int r = 0; r < 8; ++r) {
    const int e = eTile * 16 + rbase + r;
    D[(g * EE + e) * MM + mcol] = c[r];
  }
}

// --- Kernel 3: Hh[e][4(pad16),2048] = relu( Dpad[16,1024] @ wi[e][1024,2048] )
__global__ __launch_bounds__(256) void k_expert_in(const float* __restrict__ D,
                                                   const float* __restrict__ wi,
                                                   float* __restrict__ Hh) {
  const int lane = threadIdx.x & 31;
  const int wave = threadIdx.x >> 5;
  const int tile = blockIdx.x * 8 + wave;     // 64e * 128 hTiles = 8192
  const int e     = tile >> 7;
  const int hTile = tile & 127;
  const int half = lane >> 4;
  const int l15  = lane & 15;
  const int grow = l15;                        // A row: group idx, valid < 4
  const int gsafe = grow < GG ? grow : 0;      // clamp addr, zero via select
  const float amask = grow < GG ? 1.0f : 0.0f;
  const int hcol = hTile * 16 + l15;
  const float* Arow = D + (gsafe * EE + e) * MM;
  const float* Bmat = wi + (size_t)e * MM * HH;
  v8f c = {};
#pragma unroll 2
  for (int k = 0; k < MM; k += 4) {
    const int ka = k + half * 2;
    v2f a = *(const v2f*)(Arow + ka);
    a[0] *= amask;
    a[1] *= amask;
    v2f b;
    b[0] = Bmat[ka * HH + hcol];
    b[1] = Bmat[(ka + 1) * HH + hcol];
    c = wmma4(a, b, c);
  }
  const int rbase = half * 8;
#pragma unroll
  for (int r = 0; r < 8; ++r) {
    const int row = rbase + r;
    if (row < GG)
      Hh[(e * GG + row) * HH + hcol] = fmaxf(c[r], 0.0f);
  }
}

// --- Kernel 4: EO[g,e,1024] = Hhpad[16,2048] @ wo[e][2048,1024] --------------
__global__ __launch_bounds__(256) void k_expert_out(const float* __restrict__ Hh,
                                                    const float* __restrict__ wo,
                                                    float* __restrict__ EO) {
  const int lane = threadIdx.x & 31;
  const int wave = threadIdx.x >> 5;
  const int tile = blockIdx.x * 8 + wave;     // 64e * 64 mTiles = 4096
  const int e     = tile >> 6;
  const int mTile = tile & 63;
  const int half = lane >> 4;
  const int l15  = lane & 15;
  const int grow = l15;
  const int gsafe = grow < GG ? grow : 0;
  const float amask = grow < GG ? 1.0f : 0.0f;
  const int mcol = mTile * 16 + l15;
  const float* Arow = Hh + (e * GG + gsafe) * HH;
  const float* Bmat = wo + (size_t)e * HH * MM;
  v8f c = {};
#pragma unroll 2
  for (int k = 0; k < HH; k += 4) {
    const int ka = k + half * 2;
    v2f a = *(const v2f*)(Arow + ka);
    a[0] *= amask;
    a[1] *= amask;
    v2f b;
    b[0] = Bmat[ka * MM + mcol];
    b[1] = Bmat[(ka + 1) * MM + mcol];
    c = wmma4(a, b, c);
  }
  const int rbase = half * 8;
#pragma unroll
  for (int r = 0; r < 8; ++r) {
    const int row = rbase + r;
    if (row < GG)
      EO[(row * EE + e) * MM + mcol] = c[r];   // layout (G,E,M)
  }
}

// --- Kernel 5: out_g[2048,1024] = C * gates_g[2048,64] @ EO_g[64,1024] -------
__global__ __launch_bounds__(256) void k_combine(const float* __restrict__ gates,
                                                 const float* __restrict__ EO,
                                                 float* __restrict__ out) {
  const int lane = threadIdx.x & 31;
  const int wave = threadIdx.x >> 5;
  const int tile = blockIdx.x * 8 + wave;     // 4g * 128 sTiles * 64 mTiles
  const int g     = tile >> 13;
  const int sTile = (tile >> 6) & 127;
  const int mTile = tile & 63;
  const int half = lane >> 4;
  const int l15  = lane & 15;
  const int srow = sTile * 16 + l15;
  const int mcol = mTile * 16 + l15;
  v8f c = {};
#pragma unroll
  for (int k = 0; k < EE; k += 4) {           // 16 iterations
    const int ka = k + half * 2;
    v2f a = *(const v2f*)(gates + (g * SS + srow) * EE + ka); // contiguous e
    v2f b;
    b[0] = EO[(g * EE + ka) * MM + mcol];
    b[1] = EO[(g * EE + ka + 1) * MM + mcol];
    c = wmma4(a, b, c);
  }
  const int rbase = half * 8;
#pragma unroll
  for (int r = 0; r < 8; ++r)
    out[(g * SS + sTile * 16 + rbase + r) * MM + mcol] = 64.0f * c[r]; // *C
}

extern "C" void kernel_launch(void* const* d_in, const int* in_sizes, int n_in,
                              void* d_out, int out_size, void* d_ws, size_t ws_size,
                              hipStream_t stream) {
  const float* x  = (const float*)d_in[0];   // (8192, 1024)
  const float* wg = (const float*)d_in[1];   // (1024, 64)
  const float* wi = (const float*)d_in[2];   // (64, 1024, 2048)
  const float* wo = (const float*)d_in[3];   // (64, 2048, 1024)
  float* out = (float*)d_out;                // (8192, 1024)

  float* ws    = (float*)d_ws;
  float* gates = ws;                          // 8192*64   = 524288 f
  float* D     = gates + 8192 * 64;           // 4*64*1024 = 262144 f
  float* Hh    = D + 4 * 64 * 1024;           // 64*4*2048 = 524288 f
  float* EO    = Hh + 64 * 4 * 2048;          // 4*64*1024 = 262144 f  (6 MB total)

  k_gates     <<<256,  256, 0, stream>>>(x, wg, gates);
  k_dispatch  <<<128,  256, 0, stream>>>(gates, x, D);
  k_expert_in <<<1024, 256, 0, stream>>>(D, wi, Hh);
  k_expert_out<<<512,  256, 0, stream>>>(Hh, wo, EO);
  k_combine   <<<4096, 256, 0, stream>>>(gates, EO, out);
}